// Triline_33792802685225
// MI455X (gfx1250) — compile-verified
//
#include <hip/hip_runtime.h>
#include <stdint.h>

// Triline interpolation for MI455X (gfx1250).
// Store-bandwidth-bound (256MB out, ~11.5us floor at 23.3 TB/s). Tables (384KB
// f32) are staged into LDS via the CDNA5 async global->LDS copy path (ASYNCcnt)
// with a channel split (gridDim.y=2, 32 channels per block) so the 192KB stage
// fits in the 320KB WGP LDS. All 6 gathers/point are LDS b128 loads; the VMEM
// path only streams coords in and b128 non-temporal results out.

#define TPB        512
#define PTS_ITER   64          // TPB / 8 threads-per-point
#define NLINE      512
#define CH         64
#define HALF_CH    32
#define LDS_ROW    32                      // floats per row per half
#define LDS_TABLE  (NLINE * LDS_ROW)       // 16384 floats per table
#define LDS_TOTAL  (3 * LDS_TABLE)         // 49152 floats = 192 KB

typedef int   v4i __attribute__((ext_vector_type(4)));
typedef float v4f __attribute__((ext_vector_type(4)));

__device__ __forceinline__ void async_copy_b128(float* lds_dst, const float* gsrc) {
#if defined(__has_builtin) && __has_builtin(__builtin_amdgcn_global_load_async_to_lds_b128)
    __builtin_amdgcn_global_load_async_to_lds_b128(
        (__attribute__((address_space(1))) v4i*)(uintptr_t)gsrc,
        (__attribute__((address_space(3))) v4i*)lds_dst,
        0, 0);
#else
    uint32_t loff = (uint32_t)(uintptr_t)(__attribute__((address_space(3))) float*)lds_dst;
    asm volatile("global_load_async_to_lds_b128 %0, %1, off"
                 :: "v"(loff), "v"(gsrc) : "memory");
#endif
}

__device__ __forceinline__ void wait_async0() {
#if defined(__has_builtin) && __has_builtin(__builtin_amdgcn_s_wait_asynccnt)
    __builtin_amdgcn_s_wait_asynccnt(0);
#else
    asm volatile("s_wait_asynccnt 0" ::: "memory");
#endif
}

__global__ __launch_bounds__(TPB) void triline_kernel(
    const float* __restrict__ coords,
    const float* __restrict__ xl,
    const float* __restrict__ yl,
    const float* __restrict__ zl,
    const float* __restrict__ grid,
    float* __restrict__ out,
    int B)
{
    __shared__ float lds[LDS_TOTAL];

    const int tid   = threadIdx.x;
    const int half  = blockIdx.y;          // 0 -> channels [0,32), 1 -> [32,64)
    const int cbase = half * HALF_CH;

    // ---- Stage all three tables (this channel half) into LDS asynchronously ----
    // 12288 x b128 chunks; 24 per thread, tracked by ASYNCcnt.
    for (int k = tid; k < LDS_TOTAL / 4; k += TPB) {
        const int t   = k >> 12;           // table 0..2
        const int rem = k & 4095;
        const int row = rem >> 3;          // 0..511
        const int cc  = (rem & 7) << 2;    // float offset within half-row: 0,4,...,28
        const float* src = (t == 0 ? xl : (t == 1 ? yl : zl)) + row * CH + cbase + cc;
        float* dst = &lds[t * LDS_TABLE + row * LDS_ROW + cc];
        async_copy_b128(dst, src);
    }
    wait_async0();        // this wave's async copies landed in LDS
    __syncthreads();      // all waves' copies visible

    const float g0     = grid[0];
    const float inv_dg = 1.0f / (grid[1] - grid[0]);

    const int grp = tid >> 3;        // 0..63: point slot within block iteration
    const int l4  = (tid & 7) << 2;  // channel quad offset within half: 0,4,...,28

    for (int p = blockIdx.x * PTS_ITER + grp; p < B; p += gridDim.x * PTS_ITER) {
        const float c3[3] = { coords[3 * p + 0], coords[3 * p + 1], coords[3 * p + 2] };

        v4f acc = {0.0f, 0.0f, 0.0f, 0.0f};

        #pragma unroll
        for (int t = 0; t < 3; ++t) {
            float pos = (c3[t] - g0) * inv_dg;
            int i0 = (int)floorf(pos);
            i0 = i0 < 0 ? 0 : (i0 > NLINE - 2 ? NLINE - 2 : i0);
            const float w1 = pos - (float)i0;
            const float w0 = 1.0f - w1;
            const int base = t * LDS_TABLE + i0 * LDS_ROW + l4;
            const v4f a = *reinterpret_cast<const v4f*>(&lds[base]);
            const v4f b = *reinterpret_cast<const v4f*>(&lds[base + LDS_ROW]);
            acc.x = fmaf(a.x, w0, acc.x); acc.x = fmaf(b.x, w1, acc.x);
            acc.y = fmaf(a.y, w0, acc.y); acc.y = fmaf(b.y, w1, acc.y);
            acc.z = fmaf(a.z, w0, acc.z); acc.z = fmaf(b.z, w1, acc.z);
            acc.w = fmaf(a.w, w0, acc.w); acc.w = fmaf(b.w, w1, acc.w);
        }

        v4f* o = reinterpret_cast<v4f*>(&out[(size_t)p * CH + cbase + l4]);
        __builtin_nontemporal_store(acc, o);   // streamed output, never re-read
    }
}

extern "C" void kernel_launch(void* const* d_in, const int* in_sizes, int n_in,
                              void* d_out, int out_size, void* d_ws, size_t ws_size,
                              hipStream_t stream) {
    const float* coords = (const float*)d_in[0];
    const float* xl     = (const float*)d_in[1];
    const float* yl     = (const float*)d_in[2];
    const float* zl     = (const float*)d_in[3];
    const float* grid   = (const float*)d_in[4];
    float* out          = (float*)d_out;

    const int B = in_sizes[0] / 3;

    // 512 blocks total: covers any plausible WGP count at 1 block/WGP (192KB LDS),
    // while each block amortizes its 192KB stage over ~512KB of output.
    dim3 grd(256, 2, 1);   // x: point tiles (grid-stride), y: channel half
    triline_kernel<<<grd, TPB, 0, stream>>>(coords, xl, yl, zl, grid, out, B);
}